// Renderer_17059610099799
// MI455X (gfx1250) — compile-verified
//
#include <hip/hip_runtime.h>

// VolSDF-style renderer for MI455X (gfx1250, wave32).
// One ray per wave32; 8 waves per 256-thread block.
// CDNA5 paths: global_load_async_to_lds_b32 (depth staging -> LDS, ASYNCcnt),
// V_WMMA_F32_16X16X4_F32 chain implementing a triangular-matmul prefix scan.

#define NS    192
#define NM1   191
#define NFINE 64
#define WPB   8   // waves per block

typedef float v2f __attribute__((ext_vector_type(2)));
typedef float v8f __attribute__((ext_vector_type(8)));

typedef __attribute__((address_space(1))) int* as1i;
typedef __attribute__((address_space(3))) int* as3i;

__device__ __forceinline__ void async_ld_f32(const float* g, float* l) {
#if __has_builtin(__builtin_amdgcn_global_load_async_to_lds_b32)
  __builtin_amdgcn_global_load_async_to_lds_b32((as1i)g, (as3i)l, 0, 0);
#else
  unsigned lofs = (unsigned)(unsigned long long)(__attribute__((address_space(3))) float*)l;
  asm volatile("global_load_async_to_lds_b32 %0, %1, off"
               :: "v"(lofs), "v"((unsigned long long)g) : "memory");
#endif
}

__device__ __forceinline__ void wait_async0() {
#if __has_builtin(__builtin_amdgcn_s_wait_asynccnt)
  __builtin_amdgcn_s_wait_asynccnt(0);
  asm volatile("" ::: "memory");
#else
  asm volatile("s_wait_asynccnt 0" ::: "memory");
#endif
}

__device__ __forceinline__ void wait_ds0() {
  asm volatile("s_wait_dscnt 0" ::: "memory");
}

__global__ __launch_bounds__(256)
void volsdf_render(const float* __restrict__ ray,
                   const float* __restrict__ rgbs,
                   const float* __restrict__ depth,
                   const float* __restrict__ sdf,
                   const float* __restrict__ pAlpha,
                   const float* __restrict__ pBeta,
                   const float* __restrict__ bg,
                   float* __restrict__ out_rgb,
                   float* __restrict__ out_prob,
                   float* __restrict__ out_fine,
                   int nray)
{
  __shared__ float smem[WPB * 640];
  const int lane  = threadIdx.x & 31;
  const int wid   = threadIdx.x >> 5;
  const int rayId = blockIdx.x * WPB + wid;
  if (rayId >= nray) return;            // wave-uniform exit (EXEC stays all-1)

  float* wdepth = &smem[wid * 640];     // [192] depth samples
  float* wsb    = wdepth + 192;         // [256] s[j] = sigma*dintv, zero padded
  float* wcdf   = wdepth + 448;         // [192] resampling CDF

  // ---- async stage depth[192] into LDS (needed later for random access) ----
  const float* gdep = depth + (size_t)rayId * NS;
  #pragma unroll
  for (int c = 0; c < 6; ++c)
    async_ld_f32(gdep + lane + 32 * c, wdepth + lane + 32 * c);

  const float alpha = pAlpha[0];
  const float invb  = 1.0f / pBeta[0];
  const float bg0 = bg[0], bg1 = bg[1], bg2 = bg[2];
  const float rx = ray[rayId * 3 + 0], ry = ray[rayId * 3 + 1], rz = ray[rayId * 3 + 2];
  const float rl = sqrtf(rx * rx + ry * ry + rz * rz);

  // ---- coalesced sdf loads (overlap with async depth DMA) ----
  float sd[6];
  const float* gsdf = sdf + (size_t)rayId * NS;
  #pragma unroll
  for (int c = 0; c < 6; ++c) sd[c] = gsdf[lane + 32 * c];

  wait_async0();   // depth now in LDS

  // ---- s[j] = alpha*psi(sdf[j]) * (depth[j+1]-depth[j]), j<191; pad to 256 ----
  #pragma unroll
  for (int c = 0; c < 6; ++c) {
    int j = lane + 32 * c;
    float v = 0.f;
    if (j < NM1) {
      float di  = wdepth[j + 1] - wdepth[j];
      float e   = 0.5f * __expf(-fabsf(sd[c]) * invb);
      float psi = (sd[c] >= 0.f) ? e : (1.f - e);
      v = alpha * psi * di;
    }
    wsb[j] = v;
  }
  wsb[192 + lane] = 0.f;
  wsb[224 + lane] = 0.f;
  wait_ds0();      // make s[] visible across lanes

  const int mn = lane & 15;   // column (and A-row) index
  const int hf = lane >> 4;   // half-wave select

  // ---- within-chunk inclusive prefix via WMMA: D = X(16x16) * U(16x16 upper-tri)
  // decomposed into four 16x16x4 f32 WMMAs chained through C.
  // A layout: VGPR r, lane L -> A[m=L%16, k=r+2*(L/16)]
  // B layout: VGPR r, lane L -> B[k=r+2*(L/16), n=L%16]
  // D layout: VGPR r, lane L -> D[m=r+8*(L/16), n=L%16]
  v8f acc = {0.f, 0.f, 0.f, 0.f, 0.f, 0.f, 0.f, 0.f};
#if __has_builtin(__builtin_amdgcn_wmma_f32_16x16x4_f32)
  {
    const int k0 = 2 * hf, k1 = 2 * hf + 1;
    #pragma unroll
    for (int c = 0; c < 4; ++c) {
      int kc = 4 * c;
      v2f a, b;
      a.x = wsb[mn * 16 + kc + k0];
      a.y = wsb[mn * 16 + kc + k1];
      b.x = (kc + k0 <= mn) ? 1.f : 0.f;   // upper-triangular ones
      b.y = (kc + k1 <= mn) ? 1.f : 0.f;
      acc = __builtin_amdgcn_wmma_f32_16x16x4_f32(false, a, false, b,
                                                  (short)0, acc, false, false);
    }
  }
#else
  // scalar fallback: same D layout
  #pragma unroll
  for (int r = 0; r < 8; ++r) {
    int m = r + 8 * hf;
    float w = 0.f;
    for (int k = 0; k <= mn; ++k) w += wsb[m * 16 + k];
    acc[r] = w;
  }
#endif

  // ---- carry fix-up across the 12 (padded 16) chunks ----
  float tot[16], carry[16];
  #pragma unroll
  for (int r = 0; r < 8; ++r) {
    tot[r]     = __shfl(acc[r], 15, 32);   // W[r][15]
    tot[r + 8] = __shfl(acc[r], 31, 32);   // W[r+8][15]
  }
  carry[0] = 0.f;
  #pragma unroll
  for (int m = 1; m < 16; ++m) carry[m] = carry[m - 1] + tot[m - 1];

  // ---- per-sample epilogue in D layout ----
  float accR = 0.f, accG = 0.f, accB = 0.f, accO = 0.f;
  #pragma unroll
  for (int r = 0; r < 8; ++r) {
    int   m  = r + 8 * hf;
    int   j  = m * 16 + mn;
    float Ci = acc[r] + (hf ? carry[r + 8] : carry[r]);   // inclusive cumsum of s
    float s  = wsb[j];
    if (j < NM1) {
      // prob = T*a = exp(-rl*Cexcl) - exp(-rl*Cincl)
      float eX = __expf(-rl * (Ci - s));
      float eI = __expf(-rl * Ci);
      float p  = eX - eI;
      out_prob[(size_t)rayId * NM1 + j] = p;
      const float* rs = rgbs + ((size_t)rayId * NS + (size_t)j) * 3;
      accR = fmaf(p, rs[0], accR);
      accG = fmaf(p, rs[1], accG);
      accB = fmaf(p, rs[2], accB);
      accO += p;
    }
    if (j <= NM1 - 2)                     // cdf[k]=1-exp(-C[k-2]), k=j+2 <= 191
      wcdf[j + 2] = 1.f - __expf(-Ci);
  }
  if (lane < 2) wcdf[lane] = 0.f;         // cdf[0]=cdf[1]=0

  // ---- cross-lane reduction for rgb / opacity ----
  #pragma unroll
  for (int off = 16; off > 0; off >>= 1) {
    accR += __shfl_xor(accR, off, 32);
    accG += __shfl_xor(accG, off, 32);
    accB += __shfl_xor(accB, off, 32);
    accO += __shfl_xor(accO, off, 32);
  }
  if (lane == 0) {
    float w = 1.f - accO;
    out_rgb[rayId * 3 + 0] = accR + w * bg0;
    out_rgb[rayId * 3 + 1] = accG + w * bg1;
    out_rgb[rayId * 3 + 2] = accB + w * bg2;
  }

  wait_ds0();      // cdf fully visible

  // ---- inverse-CDF fine resampling: 2 fine samples per lane ----
  #pragma unroll
  for (int t = 0; t < 2; ++t) {
    int   f = lane + 32 * t;
    float u = (f + 0.5f) * (1.0f / (float)NFINE);
    int lo = 0, hi = NS;                  // lower_bound over wcdf[0..191]
    while (lo < hi) {
      int mid = (lo + hi) >> 1;
      if (wcdf[mid] < u) lo = mid + 1; else hi = mid;
    }
    int l = lo - 1; if (l < 0) l = 0;
    int h = lo;     if (h > NS - 1) h = NS - 1;
    float cl = wcdf[l], ch = wcdf[h];
    float tt = (u - cl) / (ch - cl + 1e-8f);
    float dl = wdepth[l], dh = wdepth[h];
    out_fine[(size_t)rayId * NFINE + f] = fmaf(tt, dh - dl, dl);
  }
}

extern "C" void kernel_launch(void* const* d_in, const int* in_sizes, int n_in,
                              void* d_out, int out_size, void* d_ws, size_t ws_size,
                              hipStream_t stream) {
  (void)n_in; (void)out_size; (void)d_ws; (void)ws_size;
  const float* ray   = (const float*)d_in[0];
  const float* rgbs  = (const float*)d_in[1];
  const float* depth = (const float*)d_in[2];
  const float* sdf   = (const float*)d_in[3];
  const float* alpha = (const float*)d_in[4];
  const float* beta  = (const float*)d_in[5];
  const float* bg    = (const float*)d_in[6];
  // d_in[7] = n_fine (fixed at 64, matching the reference constants)

  const int nray = in_sizes[0] / 3;            // B*HW = 32768
  float* out      = (float*)d_out;
  float* out_rgb  = out;                                    // [nray,3]
  float* out_prob = out + (size_t)nray * 3;                 // [nray,191]
  float* out_fine = out_prob + (size_t)nray * NM1;          // [nray,64]

  dim3 block(32 * WPB);
  dim3 grid((nray + WPB - 1) / WPB);
  volsdf_render<<<grid, block, 0, stream>>>(ray, rgbs, depth, sdf, alpha, beta, bg,
                                            out_rgb, out_prob, out_fine, nray);
}